// ScaledDotProductAttention_50929722196900
// MI455X (gfx1250) — compile-verified
//
#include <hip/hip_runtime.h>
#include <hip/hip_bf16.h>

typedef __attribute__((ext_vector_type(2))) float v2f;
typedef __attribute__((ext_vector_type(4))) float v4f;
typedef __attribute__((ext_vector_type(8))) float v8f;

#define Bc 2
#define Hc 8
#define Sc 2048
#define Dc 64
#define NEGV (-1e30f)
#define ROWSTRIDE 2064   // 2048 + 16 floats pad; multiple of 4 -> rows stay 16B aligned

// One workgroup = 8 waves (wave32) handles a 16-row Q tile for one (b,h).
// Phase 1: each wave computes 16x16 score tiles with V_WMMA_F32_16X16X4_F32,
//          applies 1/scale, additive bias (heads-last layout) and key mask,
//          stores rows into LDS (16 x 2064 f32 = 132KB, fits 320KB WGP LDS).
// Phase 2: each wave softmaxes 2 rows (2048 wide) with wave32 shfl reductions,
//          all LDS traffic and HBM writeback vectorized as float4 (b128).
__global__ __launch_bounds__(256) void attn_weights_wmma_kernel(
    const float* __restrict__ q,
    const float* __restrict__ kmat,
    const float* __restrict__ scale,
    const unsigned char* __restrict__ mask,   // [B,1,1,S] bool, 1 = masked
    const float* __restrict__ bias,           // [B,Sq,Skv,H] heads-last
    float* __restrict__ out)                  // [B,H,Sq,Skv]
{
    extern __shared__ float smem[];           // 16 * ROWSTRIDE floats

    const int qt   = blockIdx.x;              // q tile index (0..127)
    const int h    = blockIdx.y;
    const int b    = blockIdx.z;
    const int q0   = qt * 16;
    const int tid  = threadIdx.x;
    const int lane = tid & 31;
    const int wave = tid >> 5;
    const int half = lane >> 4;               // 0: lanes 0-15, 1: lanes 16-31
    const int lrow = lane & 15;

    const float inv_scale = 1.0f / scale[0];

    // ---- Preload A (Q tile, 16x64 f32) per ISA 16x4xK=4 layout ----
    // lane<16: row=lane, K pair {4kk+0,4kk+1}; lane>=16: row=lane-16, {4kk+2,4kk+3}
    v2f a[16];
    const size_t qbase = (((size_t)b * Hc + h) * Sc + (size_t)(q0 + lrow)) * Dc + 2 * half;
    #pragma unroll
    for (int kk = 0; kk < 16; ++kk)
        a[kk] = *(const v2f*)(q + qbase + 4 * kk);

    const size_t kmatbase = (((size_t)b * Hc + h) * Sc) * Dc;
    const unsigned char* mrow = mask + (size_t)b * Sc;

    // ---- Phase 1: score tiles ----
    for (int t = wave; t < Sc / 16; t += 8) {
        const int k0 = t * 16;

        // B tile (K^T, 4x16 per step) — same per-lane gather pattern as A.
        const size_t kb = kmatbase + (size_t)(k0 + lrow) * Dc + 2 * half;
        v2f bb[16];
        #pragma unroll
        for (int kk = 0; kk < 16; ++kk)
            bb[kk] = *(const v2f*)(kmat + kb + 4 * kk);

        // Prefetch next K tile this wave will touch (k0 + 128 rows ahead).
        if (t + 8 < Sc / 16)
            __builtin_prefetch(kmat + kb + (size_t)128 * Dc, 0, 0);

        v8f c = {};
        #pragma unroll
        for (int kk = 0; kk < 16; ++kk)
            c = __builtin_amdgcn_wmma_f32_16x16x4_f32(
                    false, a[kk], false, bb[kk], (short)0, c, false, false);

        // Post-process: scale, bias (heads-last gather), mask; stage into LDS.
        // Masked lanes skip their bias loads entirely (divergent path is a win:
        // bias is the dominant 256MB read stream).
        const bool mcol = (mrow[k0 + lrow] != 0);
        const size_t bias0 = (((size_t)b * Sc + (size_t)q0) * Sc + (size_t)(k0 + lrow)) * Hc + h;
        #pragma unroll
        for (int e = 0; e < 8; ++e) {
            const int M = e + 8 * half;       // C/D layout: VGPR e -> row e (+8 for hi half)
            float val;
            if (mcol) {
                val = NEGV;
            } else {
                val = c[e] * inv_scale + bias[bias0 + (size_t)M * (Sc * Hc)];
            }
            smem[M * ROWSTRIDE + k0 + lrow] = val;
        }
    }

    __syncthreads();

    // ---- Phase 2: row softmax (wave w owns rows 2w, 2w+1), float4 vectorized ----
    #pragma unroll
    for (int rr = 0; rr < 2; ++rr) {
        const int r = 2 * wave + rr;
        float* row = smem + r * ROWSTRIDE;

        // 1) row max (b128 LDS sweeps: 16 iters of 512B per wave)
        float m = -3.402823466e38f;
        for (int i = lane * 4; i < Sc; i += 128) {
            v4f x = *(const v4f*)(row + i);
            m = fmaxf(m, fmaxf(fmaxf(x.x, x.y), fmaxf(x.z, x.w)));
        }
        #pragma unroll
        for (int off = 16; off > 0; off >>= 1)
            m = fmaxf(m, __shfl_xor(m, off, 32));

        // 2) exp + running sum, exp values staged back into LDS
        float s = 0.0f;
        for (int i = lane * 4; i < Sc; i += 128) {
            v4f x = *(const v4f*)(row + i);
            x.x = expf(x.x - m);              // masked (-1e30) underflows to exact 0
            x.y = expf(x.y - m);
            x.z = expf(x.z - m);
            x.w = expf(x.w - m);
            *(v4f*)(row + i) = x;
            s += (x.x + x.y) + (x.z + x.w);
        }
        #pragma unroll
        for (int off = 16; off > 0; off >>= 1)
            s += __shfl_xor(s, off, 32);

        // 3) normalize + coalesced b128 writeback (512B per store instruction)
        const float inv = 1.0f / s;
        float* orow = out + ((((size_t)b * Hc + h) * Sc) + (size_t)(q0 + r)) * Sc;
        for (int i = lane * 4; i < Sc; i += 128) {
            v4f x = *(const v4f*)(row + i);
            x.x *= inv; x.y *= inv; x.z *= inv; x.w *= inv;
            *(v4f*)(orow + i) = x;
        }
    }
}

extern "C" void kernel_launch(void* const* d_in, const int* in_sizes, int n_in,
                              void* d_out, int out_size, void* d_ws, size_t ws_size,
                              hipStream_t stream) {
    (void)in_sizes; (void)n_in; (void)out_size; (void)d_ws; (void)ws_size;

    const float*         q     = (const float*)d_in[0];
    const float*         kmat  = (const float*)d_in[1];
    const float*         scale = (const float*)d_in[2];
    const unsigned char* mask  = (const unsigned char*)d_in[3];
    const float*         bias  = (const float*)d_in[4];
    float*               out   = (float*)d_out;

    const size_t shmem = (size_t)16 * ROWSTRIDE * sizeof(float); // 132 KB < 320 KB WGP LDS
    (void)hipFuncSetAttribute((const void*)attn_weights_wmma_kernel,
                              hipFuncAttributeMaxDynamicSharedMemorySize,
                              (int)shmem);

    dim3 grid(Sc / 16, Hc, Bc);  // 128 x 8 x 2 workgroups
    dim3 block(256);             // 8 wave32 waves
    attn_weights_wmma_kernel<<<grid, block, shmem, stream>>>(
        q, kmat, scale, mask, bias, out);
}